// VggStyleEncoder_44117904064998
// MI455X (gfx1250) — compile-verified
//
#include <hip/hip_runtime.h>

typedef float v2f __attribute__((ext_vector_type(2)));
typedef float v8f __attribute__((ext_vector_type(8)));

#define BS      32
#define CS      512
#define SSEG    8
#define HS      64
#define WSZ     64
#define SPATIAL (HS * WSZ)        // 4096
#define NSTEP   (SPATIAL / 4)     // 1024 WMMA steps (K=4 each)
#define SEMW    256

// ---------------------------------------------------------------------------
// Stage 1: build A-matrix (mask) in WMMA f32 16x4 lane layout, per batch/step.
// Lane l: row M = l&15 (segment; 8 = all-ones global row; 9..15 = zero pad),
//          khalf = l>>4 selects K pair {2*khalf, 2*khalf+1} within the 4-step.
// Apre[((b*NSTEP + t)*32 + lane)] = float2{mask(seg,4t+2kh), mask(seg,4t+2kh+1)}
// ---------------------------------------------------------------------------
__global__ void build_a(const int* __restrict__ sem, float* __restrict__ Apre) {
    unsigned gid  = blockIdx.x * blockDim.x + threadIdx.x;  // (b*1024+t)*32+lane
    unsigned lane = gid & 31u;
    unsigned t    = (gid >> 5) & (NSTEP - 1);
    unsigned b    = gid >> 15;
    unsigned seg  = lane & 15u;
    unsigned kh   = lane >> 4;
    unsigned pos0 = 4u * t + 2u * kh;

    v2f a;
    if (seg < SSEG) {
        // nearest downsample: segmap[b,s,h,w] = sem[b, s, 4h, 4w]
        unsigned base = (b * SSEG + seg) * (SEMW * SEMW);
        #pragma unroll
        for (int i = 0; i < 2; ++i) {
            unsigned pos = pos0 + i;
            unsigned h = pos >> 6, w = pos & 63u;
            int v = sem[base + (h * 4u) * SEMW + (w * 4u)];
            a[i] = (v != 0) ? 1.0f : 0.0f;
        }
    } else if (seg == SSEG) {
        a.x = 1.0f; a.y = 1.0f;           // global-mean row: all ones
    } else {
        a.x = 0.0f; a.y = 0.0f;           // padding rows
    }
    ((v2f*)Apre)[gid] = a;
}

// ---------------------------------------------------------------------------
// Stage 1b: area[b][s] = number of non-zero mask entries (direct write, no
// atomics, so the 0xAA-poisoned workspace needs no zero-init).
// ---------------------------------------------------------------------------
__global__ void area_kernel(const int* __restrict__ sem, float* __restrict__ area) {
    unsigned bs = blockIdx.x;             // b*8 + s
    __shared__ float red[256];
    unsigned base = bs * (SEMW * SEMW);
    float acc = 0.0f;
    for (unsigned pos = threadIdx.x; pos < SPATIAL; pos += 256u) {
        unsigned h = pos >> 6, w = pos & 63u;
        acc += (sem[base + (h * 4u) * SEMW + (w * 4u)] != 0) ? 1.0f : 0.0f;
    }
    red[threadIdx.x] = acc;
    __syncthreads();
    for (unsigned off = 128; off > 0; off >>= 1) {
        if (threadIdx.x < off) red[threadIdx.x] += red[threadIdx.x + off];
        __syncthreads();
    }
    if (threadIdx.x == 0) area[bs] = red[0];
}

// ---------------------------------------------------------------------------
// Stage 2: bandwidth-bound masked segment reduce via V_WMMA_F32_16X16X4_F32.
// Block = 128 threads = 4 waves; each wave owns one (batch, 16-channel group).
// B matrix (4x16 K x N, f32): lane l holds column N = l&15, K pair per l>>4
// -> each lane streams a float2 of consecutive spatial positions of its
// channel (non-temporal: 268 MB streamed exactly once, bigger than L2).
// C layout (16x16 f32): VGPR j, lanes 0-15 -> row j, lanes 16-31 -> row j+8.
// Row 8 (lanes 16-31, c[0]) is the global sum -> shuffled over for fallback.
// ---------------------------------------------------------------------------
__global__ void __launch_bounds__(128) seg_reduce(const float* __restrict__ codes,
                                                  const float* __restrict__ Apre,
                                                  const float* __restrict__ area,
                                                  float* __restrict__ out) {
    const unsigned wave = threadIdx.x >> 5;
    const unsigned lane = threadIdx.x & 31u;
    const unsigned b    = blockIdx.x >> 3;
    const unsigned cg   = (blockIdx.x & 7u) * 4u + wave;   // channel group 0..31
    const unsigned n    = lane & 15u;
    const unsigned kh   = lane >> 4;

    // lane's B column: channel cg*16+n, starting at spatial position 2*kh
    const v2f* __restrict__ bptr =
        (const v2f*)(codes + ((size_t)(b * CS + cg * 16u + n)) * SPATIAL) + kh;
    // lane's A slice (coalesced, L2-resident; reused by 32 waves per batch)
    const v2f* __restrict__ aptr =
        (const v2f*)Apre + (size_t)b * NSTEP * 32u + lane;

    v8f c = {0.f, 0.f, 0.f, 0.f, 0.f, 0.f, 0.f, 0.f};

    #pragma unroll 4
    for (int t = 0; t < NSTEP; ++t) {
        v2f bvec = __builtin_nontemporal_load(bptr + 2 * t);   // stream-once
        v2f avec = aptr[t * 32];
        c = __builtin_amdgcn_wmma_f32_16x16x4_f32(
                /*neg_a=*/false, avec, /*neg_b=*/false, bvec,
                /*c_mod=*/(short)0, c, /*reuse_a=*/false, /*reuse_b=*/false);
    }

    // global sum for channel n lives in lane 16+n, VGPR0 (row M=8)
    float glob = __shfl(c[0], (int)(16u + n), 32);

    if (lane < 16u) {
        const float ginv = glob * (1.0f / (float)SPATIAL);
        float* o = out + (size_t)b * (SSEG * CS) + cg * 16u + n;
        #pragma unroll
        for (int s = 0; s < SSEG; ++s) {
            float ar = area[b * SSEG + s];
            o[(size_t)s * CS] = (ar > 0.0f) ? (c[s] / ar) : ginv;
        }
    }
}

// ---------------------------------------------------------------------------
extern "C" void kernel_launch(void* const* d_in, const int* in_sizes, int n_in,
                              void* d_out, int out_size, void* d_ws, size_t ws_size,
                              hipStream_t stream) {
    const float* codes = (const float*)d_in[0];   // [32,512,64,64] f32
    const int*   sem   = (const int*)d_in[1];     // [32,8,256,256] i32
    float*       out   = (float*)d_out;           // [32, 4096]

    float* Apre = (float*)d_ws;                                   // 8 MB
    float* area = (float*)((char*)d_ws +
                           (size_t)BS * NSTEP * 32 * 2 * sizeof(float));

    // Stage 1: A-matrix precompute — 1,048,576 threads
    hipLaunchKernelGGL(build_a, dim3((BS * NSTEP * 32) / 256), dim3(256),
                       0, stream, sem, Apre);
    // Stage 1b: per-(b,s) mask areas
    hipLaunchKernelGGL(area_kernel, dim3(BS * SSEG), dim3(256),
                       0, stream, sem, area);
    // Stage 2: WMMA masked segment reduction (HBM-bound, ~11.5 us floor)
    hipLaunchKernelGGL(seg_reduce, dim3(BS * 8), dim3(128),
                       0, stream, codes, Apre, area, out);
}